// MultiScaleTriplane_51342039056740
// MI455X (gfx1250) — compile-verified
//
#include <hip/hip_runtime.h>

typedef float v4f __attribute__((ext_vector_type(4)));

#define C_CH 32

// ---------------------------------------------------------------------------
// Per-dimension bilinear setup, mirroring the reference exactly:
//   ix = t * (R-1);  x0 = floor(ix);  wx1 = ix - x0;  wx0 = 1 - wx1
// Corners fetched as the pair (xb, xb+1) with xb clamped to [0, R-2]; weights
// are shifted/zeroed so out-of-range corners contribute exactly 0 (zeros pad).
// ---------------------------------------------------------------------------
struct DimS { int b; float wl, wh; };

__device__ __forceinline__ DimS dimSetup(float t, int R) {
  float ix = t * (float)(R - 1);
  float fl = floorf(ix);
  float fr = ix - fl;          // wx1
  float w0 = 1.0f - fr;        // wx0
  int x0 = (int)fl;
  int xb = x0 < 0 ? 0 : (x0 > R - 2 ? R - 2 : x0);
  DimS d;
  d.b  = xb;
  d.wl = (xb == x0) ? w0 : ((xb == x0 + 1) ? fr : 0.0f);
  d.wh = (xb == x0) ? fr : ((xb == x0 - 1) ? w0 : 0.0f);
  return d;
}

// ---------------------------------------------------------------------------
// Packed-layout sampler: planes repacked as [plane][H][W][C] so each corner's
// 32 channels are 128B contiguous. Wave32 handles 4 points: lane = psub*8+cg,
// each lane owns 4 channels (one B128 per corner, corners via imm offsets).
// ---------------------------------------------------------------------------
template <int R>
__device__ __forceinline__ void planeSampleP(const float* __restrict__ P, int cg4,
                                             const DimS& X, const DimS& Y, v4f& acc) {
  const float* q = P + ((Y.b * R + X.b) * C_CH + cg4);
  v4f v00 = *(const v4f*)(q);                    // (y0, x0)
  v4f v01 = *(const v4f*)(q + C_CH);             // (y0, x0+1)  imm +128B
  v4f v10 = *(const v4f*)(q + R * C_CH);         // (y0+1, x0)  imm +R*128B
  v4f v11 = *(const v4f*)(q + R * C_CH + C_CH);  // (y0+1, x0+1)
  float w00 = X.wl * Y.wl, w01 = X.wh * Y.wl;
  float w10 = X.wl * Y.wh, w11 = X.wh * Y.wh;
  acc += v00 * w00;
  acc += v01 * w01;
  acc += v10 * w10;
  acc += v11 * w11;
}

template <int R>
__device__ __forceinline__ void sampleScaleP(const float* __restrict__ base,
                                             float tx, float ty, float tz,
                                             int cg4, float* __restrict__ outp) {
  DimS dx = dimSetup(tx, R);
  DimS dy = dimSetup(ty, R);
  DimS dz = dimSetup(tz, R);
  v4f acc = {0.0f, 0.0f, 0.0f, 0.0f};
  planeSampleP<R>(base,                    cg4, dx, dy, acc);  // plane0: (x, y)
  planeSampleP<R>(base + R * R * C_CH,     cg4, dy, dz, acc);  // plane1: (y, z)
  planeSampleP<R>(base + 2 * R * R * C_CH, cg4, dx, dz, acc);  // plane2: (x, z)
  *(v4f*)(outp + cg4) = acc;
}

__global__ __launch_bounds__(256) void sample_packed(
    const float* __restrict__ coords, const float* __restrict__ wsBase,
    float* __restrict__ out, int npg) {
  const int lane = threadIdx.x & 31;
  const int wic  = threadIdx.x >> 5;
  const int wid  = __builtin_amdgcn_readfirstlane(
      (int)(blockIdx.x * (blockDim.x >> 5) + wic));
  const int nw   = (int)(gridDim.x * (blockDim.x >> 5));
  const int psub = lane >> 3;        // point within group of 4
  const int cg4  = (lane & 7) << 2;  // first of this lane's 4 channels

  const float* s0 = wsBase;
  const float* s1 = s0 + (size_t)3 * 256 * 256 * C_CH;
  const float* s2 = s1 + (size_t)3 * 128 * 128 * C_CH;

  for (int pg = wid; pg < npg; pg += nw) {
    const int p = (pg << 2) + psub;
    // gfx1250 global_prefetch_b8 of next iteration's coords (speculative: safe OOB)
    __builtin_prefetch(coords + (size_t)((pg + nw) << 2) * 3, 0, 1);

    const float cx = coords[(size_t)p * 3 + 0];
    const float cy = coords[(size_t)p * 3 + 1];
    const float cz = coords[(size_t)p * 3 + 2];
    // t = ((coord+1)/2 + 1) * 0.5 ; exact op order of the reference
    const float tx = ((cx + 1.0f) * 0.5f + 1.0f) * 0.5f;
    const float ty = ((cy + 1.0f) * 0.5f + 1.0f) * 0.5f;
    const float tz = ((cz + 1.0f) * 0.5f + 1.0f) * 0.5f;

    float* op = out + (size_t)p * 96;
    sampleScaleP<256>(s0, tx, ty, tz, cg4, op);
    sampleScaleP<128>(s1, tx, ty, tz, cg4, op + 32);
    sampleScaleP<64 >(s2, tx, ty, tz, cg4, op + 64);
  }
}

// ---------------------------------------------------------------------------
// Repack [3][C][R][R] -> [3][R][R][C] with an LDS 32x32 transpose tile.
// Load phase uses the CDNA5 async global->LDS pipe (GLOBAL_LOAD_ASYNC_TO_LDS_B32,
// tracked by ASYNCcnt) instead of a VGPR round-trip; completion is enforced
// with s_wait_asynccnt 0 before the workgroup barrier. Reads and writes are
// both 128B coalesced. Grid: 3 * R * (R/32) blocks of 256 threads.
// ---------------------------------------------------------------------------
template <int R>
__global__ __launch_bounds__(256) void repack_plane(
    const float* __restrict__ in, float* __restrict__ outp) {
  __shared__ float tile[32][33];  // +1 pad: conflict-free transposed reads
  const int XT = R / 32;
  const int bid = blockIdx.x;
  const int xt = bid % XT;
  const int y  = (bid / XT) % R;
  const int q  = bid / (XT * R);
  const int tx = threadIdx.x & 31;
  const int ty = threadIdx.x >> 5;  // 0..7

  const float* src = in + (size_t)q * C_CH * R * R + (size_t)y * R + (size_t)xt * 32;
#pragma unroll
  for (int cc = 0; cc < 4; ++cc) {
    const int c = cc * 8 + ty;
    const float* gptr = src + (size_t)c * R * R + tx;
    // Low 32 bits of the generic pointer to a __shared__ object are the
    // workgroup-relative LDS byte offset (flat LDS aperture: addr[31:0]).
    unsigned ldsOff = (unsigned)(unsigned long long)(&tile[c][tx]);
    asm volatile("global_load_async_to_lds_b32 %0, %1, off"
                 :
                 : "v"(ldsOff), "v"(gptr)
                 : "memory");
  }
  asm volatile("s_wait_asynccnt 0x0" ::: "memory");
  __syncthreads();

  float* dst = outp + (((size_t)q * R + y) * R + (size_t)xt * 32) * C_CH;
#pragma unroll
  for (int cc = 0; cc < 4; ++cc) {
    const int x = cc * 8 + ty;
    dst[(size_t)x * C_CH + tx] = tile[tx][x];  // banks (tx*33+x)%32 distinct
  }
}

// ---------------------------------------------------------------------------
// Direct-layout fallback (used only if ws_size can't hold the repacked planes,
// and for any tail points). One thread per point.
// ---------------------------------------------------------------------------
template <int R>
__device__ __forceinline__ void planeSampleD(const float* __restrict__ P,
                                             const DimS& X, const DimS& Y,
                                             float* acc) {
  const float* r0 = P + (Y.b * R + X.b);
  const float w00 = X.wl * Y.wl, w01 = X.wh * Y.wl;
  const float w10 = X.wl * Y.wh, w11 = X.wh * Y.wh;
#pragma unroll
  for (int c = 0; c < C_CH; ++c) {
    const float* q = r0 + (size_t)c * R * R;
    float a0 = q[0], a1 = q[1], b0 = q[R], b1 = q[R + 1];
    acc[c] += a0 * w00 + a1 * w01 + b0 * w10 + b1 * w11;
  }
}

template <int R>
__device__ __forceinline__ void sampleScaleD(const float* __restrict__ plane,
                                             float tx, float ty, float tz,
                                             float* __restrict__ outp) {
  DimS dx = dimSetup(tx, R);
  DimS dy = dimSetup(ty, R);
  DimS dz = dimSetup(tz, R);
  float acc[C_CH];
#pragma unroll
  for (int c = 0; c < C_CH; ++c) acc[c] = 0.0f;
  planeSampleD<R>(plane,                            dx, dy, acc);
  planeSampleD<R>(plane + (size_t)R * R * C_CH,     dy, dz, acc);
  planeSampleD<R>(plane + (size_t)2 * R * R * C_CH, dx, dz, acc);
#pragma unroll
  for (int c = 0; c < C_CH; ++c) outp[c] = acc[c];
}

__global__ __launch_bounds__(256) void sample_direct(
    const float* __restrict__ coords,
    const float* __restrict__ p0, const float* __restrict__ p1,
    const float* __restrict__ p2, float* __restrict__ out,
    int pstart, int npts) {
  const int p = pstart + (int)(blockIdx.x * blockDim.x + threadIdx.x);
  if (p >= npts) return;
  const float cx = coords[(size_t)p * 3 + 0];
  const float cy = coords[(size_t)p * 3 + 1];
  const float cz = coords[(size_t)p * 3 + 2];
  const float tx = ((cx + 1.0f) * 0.5f + 1.0f) * 0.5f;
  const float ty = ((cy + 1.0f) * 0.5f + 1.0f) * 0.5f;
  const float tz = ((cz + 1.0f) * 0.5f + 1.0f) * 0.5f;
  float* op = out + (size_t)p * 96;
  sampleScaleD<256>(p0, tx, ty, tz, op);
  sampleScaleD<128>(p1, tx, ty, tz, op + 32);
  sampleScaleD<64 >(p2, tx, ty, tz, op + 64);
}

// ---------------------------------------------------------------------------
extern "C" void kernel_launch(void* const* d_in, const int* in_sizes, int n_in,
                              void* d_out, int out_size, void* d_ws, size_t ws_size,
                              hipStream_t stream) {
  (void)n_in; (void)out_size;
  const float* coords = (const float*)d_in[0];
  const float* p0 = (const float*)d_in[1];
  const float* p1 = (const float*)d_in[2];
  const float* p2 = (const float*)d_in[3];
  float* out = (float*)d_out;
  const int npts = in_sizes[0] / 3;

  const size_t e0 = (size_t)3 * C_CH * 256 * 256;
  const size_t e1 = (size_t)3 * C_CH * 128 * 128;
  const size_t e2 = (size_t)3 * C_CH * 64 * 64;
  const size_t need = (e0 + e1 + e2) * sizeof(float);  // ~31.5 MiB

  if (ws_size >= need) {
    float* w0 = (float*)d_ws;
    float* w1 = w0 + e0;
    float* w2 = w1 + e1;
    repack_plane<256><<<3 * 256 * (256 / 32), 256, 0, stream>>>(p0, w0);
    repack_plane<128><<<3 * 128 * (128 / 32), 256, 0, stream>>>(p1, w1);
    repack_plane<64 ><<<3 * 64  * (64  / 32), 256, 0, stream>>>(p2, w2);

    const int npg = npts >> 2;
    if (npg > 0) {
      sample_packed<<<2048, 256, 0, stream>>>(coords, w0, out, npg);
    }
    const int tail = npts & 3;
    if (tail) {
      sample_direct<<<1, 256, 0, stream>>>(coords, p0, p1, p2, out, npg << 2, npts);
    }
  } else {
    sample_direct<<<(npts + 255) / 256, 256, 0, stream>>>(coords, p0, p1, p2,
                                                          out, 0, npts);
  }
}